// QuanvolutionEnhanced_65481071404468
// MI455X (gfx1250) — compile-verified
//
#include <hip/hip_runtime.h>
#include <hip/hip_bf16.h>
#include <math.h>

// D = A(16x4,f32) * B(4x16,f32) + C(16x16,f32) accumulated over 196 K-chunks.
typedef __attribute__((ext_vector_type(2))) float v2f;
typedef __attribute__((ext_vector_type(8))) float v8f;

#define IMGSZ 784
#define NCLS  10

__global__ __launch_bounds__(32)
void quanv_wmma_fused(const float* __restrict__ x,      // [2048,784]
                      const float* __restrict__ theta,  // [4]
                      const float* __restrict__ W,      // [10,784]
                      const float* __restrict__ bias,   // [10]
                      float* __restrict__ out)          // [2048,10] log-softmax
{
    const int lane = threadIdx.x;       // 0..31 (wave32)
    const int tile = blockIdx.x;        // batch rows [tile*16, tile*16+16)
    const int half = lane >> 4;         // 0 -> A holds K={0,1}; 1 -> K={2,3}
    const int r    = lane & 15;         // A row (M) / B column (N) index
    const int bidx = tile * 16 + r;     // batch row this lane feeds into A

    // theta is tiny/uniform: scalar-cached loads
    const float cT0 = __cosf(theta[0]);
    const float T1  = theta[1];
    const float cT3 = __cosf(theta[3]);   // theta[2] (RZ) provably drops out

    const bool  nvalid = (r < NCLS);
    const float* Wrow  = W + (nvalid ? r : 0) * IMGSZ;  // clamp to stay in-bounds

    v8f acc = {};                         // C/D accumulator, 8 VGPRs
    const float* xb = x + (size_t)bidx * IMGSZ;

    int p4 = 0;                           // feature index base = patch*4
    for (int pr = 0; pr < 14; ++pr) {
        const float* row0 = xb + (2 * pr) * 28;
        const float* row1 = row0 + 28;
        #pragma unroll 2
        for (int pc = 0; pc < 14; ++pc, p4 += 4) {
            // 2x2 stride-2 patch: two aligned float2 loads
            v2f px01 = *(const v2f*)(row0 + 2 * pc);
            v2f px23 = *(const v2f*)(row1 + 2 * pc);

            // closed-form per-wire Z expectations before the CNOT ring
            float u0 = __cosf(px01.x) * cT0;
            float u1 = __cosf(px01.y + T1);
            float u2 = __cosf(px23.x);
            float u3 = __cosf(px23.y) * cT3;
            float u01 = u0 * u1;

            // features of this patch: f0=u1u2u3 f1=u0u1 f2=u0u1u2 f3=u0u1u2u3
            // A fragment 16x4 f32: lanes<16 -> (K0,K1), lanes>=16 -> (K2,K3)
            v2f a;
            if (half == 0) { a.x = u1 * u2 * u3; a.y = u01; }
            else           { float u012 = u01 * u2; a.x = u012; a.y = u012 * u3; }

            // B fragment 4x16 f32 = W^T slice [p4..p4+3][n], zero-padded n>=10
            int kb = p4 + half * 2;
            v2f bf;
            bf.x = nvalid ? Wrow[kb]     : 0.0f;
            bf.y = nvalid ? Wrow[kb + 1] : 0.0f;

            acc = __builtin_amdgcn_wmma_f32_16x16x4_f32(
                false, a, false, bf, (short)0, acc, false, false);
        }
    }

    // Epilogue: D layout -> VGPR j holds row m = j + 8*half, lane's col n = r.
    // Per-row log_softmax over n=0..9 via 16-wide xor-shuffle reductions.
    #pragma unroll
    for (int j = 0; j < 8; ++j) {
        const int m = j + 8 * half;
        float logit = nvalid ? (acc[j] + bias[r]) : -3.0e38f;

        float mx = logit;
        #pragma unroll
        for (int off = 8; off >= 1; off >>= 1)
            mx = fmaxf(mx, __shfl_xor(mx, off, 16));

        float e = nvalid ? __expf(logit - mx) : 0.0f;
        float s = e;
        #pragma unroll
        for (int off = 8; off >= 1; off >>= 1)
            s += __shfl_xor(s, off, 16);

        float lse = mx + __logf(s);
        if (nvalid)
            out[(size_t)(tile * 16 + m) * NCLS + r] = logit - lse;
    }
}

extern "C" void kernel_launch(void* const* d_in, const int* in_sizes, int n_in,
                              void* d_out, int out_size, void* d_ws, size_t ws_size,
                              hipStream_t stream) {
    const float* x     = (const float*)d_in[0];   // [2048,1,28,28]
    const float* theta = (const float*)d_in[1];   // [4]
    const float* W     = (const float*)d_in[2];   // [10,784]
    const float* bias  = (const float*)d_in[3];   // [10]
    float* out = (float*)d_out;                   // [2048,10]

    // 2048 batch rows / 16 per tile = 128 waves; 1 wave32 per workgroup
    quanv_wmma_fused<<<128, 32, 0, stream>>>(x, theta, W, bias, out);
}